// GlobalAttention_41858751267187
// MI455X (gfx1250) — compile-verified
//
#include <hip/hip_runtime.h>
#include <math.h>

#define B_ 32
#define N_ 1024
#define F_ 128
#define D_ 128

typedef float v2f __attribute__((ext_vector_type(2)));
typedef float v8f __attribute__((ext_vector_type(8)));

#define LDSPITCH 132   // 128 + 4 pad: row stride not multiple of 64 banks

// ---------------------------------------------------------------------------
// Kernel 0: per-batch column sum of A:  Asum[b,f] = sum_n A[b,n,f]
// ---------------------------------------------------------------------------
__global__ __launch_bounds__(128)
void asum_kernel(const float* __restrict__ A, float* __restrict__ ASUM)
{
    const int b = blockIdx.x;
    const int f = threadIdx.x;
    const float* Ab = A + (size_t)b * N_ * F_ + f;
    float s = 0.f;
    for (int n = 0; n < N_; ++n) s += Ab[(size_t)n * F_];
    ASUM[b * F_ + f] = s;
}

// ---------------------------------------------------------------------------
// Kernel 1: per-batch   Ksum = Asum@Wk + N*bk   (in LDS),
//           u = Wq @ Ksum,   c = bq . Ksum
// ---------------------------------------------------------------------------
__global__ __launch_bounds__(128)
void prep_kernel(const float* __restrict__ ASUM,
                 const float* __restrict__ Wq, const float* __restrict__ bq,
                 const float* __restrict__ Wk, const float* __restrict__ bk,
                 float* __restrict__ U, float* __restrict__ C)
{
    __shared__ float ksl[D_];
    const int b = blockIdx.x;
    const int t = threadIdx.x;
    const float* as = ASUM + b * F_;

    // Ksum[d] = sum_f Asum[f] * Wk[f,d] + N*bk[d]
    float s = (float)N_ * bk[t];
    for (int f = 0; f < F_; ++f) s += as[f] * Wk[f * D_ + t];
    ksl[t] = s;
    __syncthreads();

    // u[f] = sum_d Wq[f,d] * Ksum[d]
    float u = 0.f;
    for (int d = 0; d < D_; ++d) u += Wq[t * D_ + d] * ksl[d];
    U[b * F_ + t] = u;

    if (t == 0) {
        float c = 0.f;
        for (int d = 0; d < D_; ++d) c += bq[d] * ksl[d];
        C[b] = c;
    }
}

// ---------------------------------------------------------------------------
// Kernel 2: fused WMMA projection. One LDS A-fragment feeds THREE
// v_wmma_f32_16x16x4_f32 (Q, K, V tiles). V is written out; Q/K never touch
// memory: self = q.k reduced in-register (shfl_xor over the 16-lane row
// stripe), q.Ksum folded to dk*(A.u + c) from the LDS tile, and the final
// masked agg row-sum AGG[b,n] stored directly.
// ---------------------------------------------------------------------------
__global__ __launch_bounds__(128)
void fused_proj_kernel(const float* __restrict__ A, const float* __restrict__ mask,
                       const float* __restrict__ Wq, const float* __restrict__ bq,
                       const float* __restrict__ Wk, const float* __restrict__ bk,
                       const float* __restrict__ Wv, const float* __restrict__ bv,
                       const float* __restrict__ U, const float* __restrict__ C,
                       float* __restrict__ V, float* __restrict__ AGG, float dk)
{
    __shared__ float As[64 * LDSPITCH];   // 33.8 KB of 320 KB WGP LDS
    __shared__ float ul[F_];
    __shared__ float selfl[64];

    const int b    = blockIdx.y;
    const int rt   = blockIdx.x;          // 64-row tile
    const int tid  = threadIdx.x;
    const int wave = tid >> 5;
    const int lane = tid & 31;
    const int half = lane >> 4;
    const int l15  = lane & 15;

    const float* Abase = A + ((size_t)b * N_ + (size_t)rt * 64) * F_;

    // Prefetch hot weight matrices (global_prefetch_b8)
    for (int i = tid; i < (F_ * D_) / 16; i += 128) {
        __builtin_prefetch(Wq + i * 16, 0, 3);
        __builtin_prefetch(Wk + i * 16, 0, 3);
        __builtin_prefetch(Wv + i * 16, 0, 3);
    }

    // Stage 64x128 fp32 A-tile into LDS (padded pitch kills bank conflicts)
    for (int idx = tid * 4; idx < 64 * F_; idx += 128 * 4) {
        const float4 v4 = *(const float4*)(Abase + idx);
        const int row = idx >> 7;
        const int col = idx & 127;
        float* dst = &As[row * LDSPITCH + col];
        dst[0] = v4.x; dst[1] = v4.y; dst[2] = v4.z; dst[3] = v4.w;
    }
    ul[tid] = U[b * F_ + tid];            // tid < 128 == F_
    __syncthreads();

    const int m_local = wave * 16 + l15;      // A fragment row (M = lane&15)
    const int row0    = rt * 64 + wave * 16;  // first global row of this wave

    float p[8] = {0.f, 0.f, 0.f, 0.f, 0.f, 0.f, 0.f, 0.f};
    float* Vb = V + (size_t)b * N_ * D_;

    for (int nt = 0; nt < D_ / 16; ++nt) {
        const int n0  = nt * 16;
        const int col = n0 + l15;
        v8f qa = {0.f,0.f,0.f,0.f,0.f,0.f,0.f,0.f};
        v8f ka = {0.f,0.f,0.f,0.f,0.f,0.f,0.f,0.f};
        v8f va = {0.f,0.f,0.f,0.f,0.f,0.f,0.f,0.f};

        #pragma unroll 4
        for (int kk = 0; kk < F_; kk += 4) {
            const int kb = kk + 2 * half;     // ISA 7.12.2 A/B lane-half split
            v2f a, fq, fk, fv;
            a.x  = As[m_local * LDSPITCH + kb];
            a.y  = As[m_local * LDSPITCH + kb + 1];
            fq.x = Wq[(size_t)kb * D_ + col];  fq.y = Wq[(size_t)(kb + 1) * D_ + col];
            fk.x = Wk[(size_t)kb * D_ + col];  fk.y = Wk[(size_t)(kb + 1) * D_ + col];
            fv.x = Wv[(size_t)kb * D_ + col];  fv.y = Wv[(size_t)(kb + 1) * D_ + col];
            qa = __builtin_amdgcn_wmma_f32_16x16x4_f32(false, a, false, fq, (short)0, qa, false, false);
            ka = __builtin_amdgcn_wmma_f32_16x16x4_f32(false, a, false, fk, (short)0, ka, false, false);
            va = __builtin_amdgcn_wmma_f32_16x16x4_f32(false, a, false, fv, (short)0, va, false, false);
        }

        const float bqv = bq[col], bkv = bk[col], bvv = bv[col];
        #pragma unroll
        for (int r = 0; r < 8; ++r) {         // C/D: row = r + 8*half
            const float qv = qa[r] + bqv;
            const float kv = ka[r] + bkv;
            p[r] += qv * kv;                  // partial q.k (per col stripe)
            const int row = row0 + half * 8 + r;
            Vb[(size_t)row * D_ + col] = va[r] + bvv;
        }
    }

    // Reduce q.k across the 16-lane column stripe (stays within lane-half)
    #pragma unroll
    for (int r = 0; r < 8; ++r) {
        float v = p[r];
        v += __shfl_xor(v, 1, 32);
        v += __shfl_xor(v, 2, 32);
        v += __shfl_xor(v, 4, 32);
        v += __shfl_xor(v, 8, 32);
        if (l15 == 0) selfl[wave * 16 + half * 8 + r] = dk * v;  // q scaled by dk
    }
    __syncthreads();

    // agg[row] = mask^2 * ( dk*(A_row . u + c) - self[row] )
    if (tid < 64) {
        const int row = tid;
        const float* ar = &As[row * LDSPITCH];
        float adot = 0.f;
        for (int f = 0; f < F_; f += 4)
            adot += ar[f]   * ul[f]   + ar[f+1] * ul[f+1]
                  + ar[f+2] * ul[f+2] + ar[f+3] * ul[f+3];
        const float qdots = dk * (adot + C[b]);
        const float mi = mask[b * N_ + rt * 64 + row];
        float agg = mi * (qdots - selfl[row]);
        agg *= mi;
        AGG[b * N_ + rt * 64 + row] = agg;
    }
}

// ---------------------------------------------------------------------------
// Kernel 3: L2-normalize agg over atoms, softmax, attn output + context.
// One 1024-thread block per batch (thread = atom row).
// ---------------------------------------------------------------------------
__global__ __launch_bounds__(1024)
void finalize_kernel(const float* __restrict__ AGG, const float* __restrict__ V,
                     const float* __restrict__ mask, float* __restrict__ out)
{
    __shared__ float red[1024];
    __shared__ float wbuf[1024];

    const int b = blockIdx.x;
    const int i = threadIdx.x;

    float agg = AGG[b * N_ + i];
    const float mi = mask[b * N_ + i];

    // nrm = sqrt(sum_i agg^2)
    red[i] = agg * agg;
    __syncthreads();
    for (int s = 512; s > 0; s >>= 1) {
        if (i < s) red[i] += red[i + s];
        __syncthreads();
    }
    const float nrm = sqrtf(red[0]);
    __syncthreads();

    agg = agg / nrm + (1.0f - mi) * (-1e9f);

    // softmax over atom axis
    red[i] = agg;
    __syncthreads();
    for (int s = 512; s > 0; s >>= 1) {
        if (i < s) red[i] = fmaxf(red[i], red[i + s]);
        __syncthreads();
    }
    const float mx = red[0];
    __syncthreads();
    const float e = expf(agg - mx);
    red[i] = e;
    __syncthreads();
    for (int s = 512; s > 0; s >>= 1) {
        if (i < s) red[i] += red[i + s];
        __syncthreads();
    }
    const float attn = e / red[0];
    __syncthreads();

    out[b * N_ + i] = attn;            // attn [B,N,1]
    wbuf[i] = mi * attn;
    __syncthreads();

    // context[d] = sum_i wbuf[i] * V[b,i,d]; 8 row-chunks in parallel
    const int d     = i & 127;
    const int chunk = i >> 7;
    const float* Vb = V + (size_t)b * N_ * D_;
    float part = 0.f;
    for (int r = chunk * 128; r < chunk * 128 + 128; ++r)
        part += wbuf[r] * Vb[(size_t)r * D_ + d];
    red[chunk * 128 + d] = part;
    __syncthreads();
    if (i < 128) {
        float s = 0.f;
        #pragma unroll
        for (int c = 0; c < 8; ++c) s += red[c * 128 + i];
        out[B_ * N_ + b * D_ + i] = s; // context [B,D]
    }
}

// ---------------------------------------------------------------------------
extern "C" void kernel_launch(void* const* d_in, const int* in_sizes, int n_in,
                              void* d_out, int out_size, void* d_ws, size_t ws_size,
                              hipStream_t stream)
{
    (void)in_sizes; (void)n_in; (void)out_size; (void)ws_size;

    const float* A    = (const float*)d_in[0];
    const float* mask = (const float*)d_in[1];
    const float* Wq   = (const float*)d_in[2];
    const float* bq   = (const float*)d_in[3];
    const float* Wk   = (const float*)d_in[4];
    const float* bk   = (const float*)d_in[5];
    const float* Wv   = (const float*)d_in[6];
    const float* bv   = (const float*)d_in[7];
    float* out = (float*)d_out;
    float* ws  = (float*)d_ws;

    const size_t BND = (size_t)B_ * N_ * D_;
    float* Vws  = ws;                       // B*N*D
    float* ASUM = ws + BND;                 // B*F
    float* U    = ASUM + (size_t)B_ * F_;   // B*F
    float* Cc   = U + (size_t)B_ * F_;      // B
    float* AGG  = Cc + B_;                  // B*N

    const float dk = (float)(1.0 / (sqrt((double)D_) * (double)N_));

    asum_kernel<<<B_, 128, 0, stream>>>(A, ASUM);
    prep_kernel<<<B_, 128, 0, stream>>>(ASUM, Wq, bq, Wk, bk, U, Cc);
    fused_proj_kernel<<<dim3(16, 32), 128, 0, stream>>>(A, mask, Wq, bq, Wk, bk,
                                                        Wv, bv, U, Cc, Vws, AGG, dk);
    finalize_kernel<<<B_, 1024, 0, stream>>>(AGG, Vws, mask, out);
}